// Net_77687368450207
// MI455X (gfx1250) — compile-verified
//
#include <hip/hip_runtime.h>
#include <hip/hip_bf16.h>

typedef float v2f __attribute__((ext_vector_type(2)));
typedef float v8f __attribute__((ext_vector_type(8)));
typedef unsigned int v4u __attribute__((ext_vector_type(4)));
typedef int v4i __attribute__((ext_vector_type(4)));
typedef int v8i __attribute__((ext_vector_type(8)));

#define C_DIM 128
#define B_SZ  256
#define WALKN 7
#define NP_N  (B_SZ * 128)      // 32768
#define EP_N  (NP_N * 16)       // 524288
#define NN_N  128
#define EN_N  (NN_N * 16)       // 2048

// ---------------- degree / norm ----------------
__global__ void k_fill1(float* __restrict__ p, int n) {
    int i = blockIdx.x * blockDim.x + threadIdx.x;
    if (i < n) p[i] = 1.0f;                     // self-loop contributes 1 to degree
}

__global__ void k_degcount(const int* __restrict__ dst, float* __restrict__ deg, int E) {
    int e = blockIdx.x * blockDim.x + threadIdx.x;
    if (e < E) atomicAdd(&deg[dst[e]], 1.0f);
}

__global__ void k_rsqrt_inplace(float* __restrict__ p, int n) {
    int i = blockIdx.x * blockDim.x + threadIdx.x;
    if (i < n) p[i] = rsqrtf(p[i]);             // deg >= 1 always (self loop)
}

// ---------------- fp32 WMMA GEMM: O[N x 128] = H[N x 128] @ W[128 x 128] ----------------
// Block = 256 threads = 8 waves; block covers a 128-row strip. Each wave computes a
// full 16x128 output strip (8 column tiles, 8 v8f accumulators), so each A-fragment
// is reused across 8 WMMAs and the LDS-staged W is amortized over 128 rows.
// W (64KB) is DMA'd into LDS by the Tensor Data Mover (one tensor_load_to_lds from
// wave 0), tracked by TENSORcnt. The address of Ws is escaped into an asm with a
// memory clobber so alias analysis must assume it was written (the TDM descriptor
// carries the LDS address only as an integer, invisible to the compiler).
// A (16x4 f32) layout: lane L holds M=L%16; VGPR0 = K = (L<16?0:2), VGPR1 = K+1.
// B (4x16 f32) mirrored; C/D (16x16 f32): VGPR r -> row r (+8 for lanes 16-31), col = lane%16.
__global__ __launch_bounds__(256) void k_gemm_wmma(const float* __restrict__ H,
                                                   const float* __restrict__ W,
                                                   float* __restrict__ O) {
    __shared__ float Ws[C_DIM * C_DIM];   // 64 KB, at LDS offset 0 (only shared object)

    if ((threadIdx.x >> 5) == 0) {
        // Tensor DMA descriptor (D#) per CDNA5 ISA ch.8: 2D tensor, 4-byte elements,
        // tensor 128x128, tile 128x128, dim0 stride 128, dest LDS offset 0.
        unsigned long long ga = (unsigned long long)(const void*)W;
        v4u g0;
        g0[0] = 1u;                                    // count=1 (valid), user mode
        g0[1] = 0u;                                    // lds_addr = 0 (Ws offset)
        g0[2] = (unsigned int)ga;                      // global_addr[31:0]
        g0[3] = (unsigned int)((ga >> 32) & 0x01FFFFFFu) | (2u << 30);  // addr[56:32] | type=2
        v8i g1;
        g1[0] = (int)(2u << 16);                       // workgroup_mask=0, data_size=2 (4B)
        g1[1] = (int)(128u << 16);                     // tensor_dim0[15:0] at bits[63:48]
        g1[2] = (int)(128u << 16);                     // tensor_dim1[15:0] at bits[95:80]
        g1[3] = (int)(128u << 16);                     // tile_dim0 at bits[127:112]
        g1[4] = 128;                                   // tile_dim1 (tile_dim2 = 0)
        g1[5] = 128;                                   // tensor_dim0_stride[31:0]
        g1[6] = 0;                                     // stride0 hi / tensor_dim1_stride lo
        g1[7] = 0;
        v4i gz4 = {};                                  // groups 2/3 unused for 2D tensor
        v8i gz8 = {};                                  // 6-arg toolchain extra group (zero-fill)
        __builtin_amdgcn_tensor_load_to_lds(g0, g1, gz4, gz4, gz8, 0);
        __builtin_amdgcn_s_wait_tensorcnt(0);          // only issuing wave tracks TENSORcnt
    }
    // Escape Ws's address into an asm that clobbers memory: without the capture,
    // LLVM sees an LDS global with no stores and folds all its loads to undef.
    {
        const float* esc = &Ws[0];
        asm volatile("" :: "v"(esc) : "memory");
    }
    __syncthreads();

    const int waveid = threadIdx.x >> 5;
    const int lane   = threadIdx.x & 31;
    const int nlane  = lane & 15;
    const int koff   = (lane >> 4) << 1;               // 0 for lanes 0-15, 2 for lanes 16-31
    const int mrow   = blockIdx.x * 128 + waveid * 16 + nlane;
    const float* hrow = H + (size_t)mrow * C_DIM;

    v8f zero = {};
    v8f acc[8];
#pragma unroll
    for (int t = 0; t < 8; ++t) acc[t] = zero;

#pragma unroll 4
    for (int k0 = 0; k0 < C_DIM; k0 += 4) {
        v2f a;
        a.x = hrow[k0 + koff];
        a.y = hrow[k0 + koff + 1];
#pragma unroll
        for (int t = 0; t < 8; ++t) {
            v2f b;
            b.x = Ws[(k0 + koff) * C_DIM + t * 16 + nlane];
            b.y = Ws[(k0 + koff + 1) * C_DIM + t * 16 + nlane];
            acc[t] = __builtin_amdgcn_wmma_f32_16x16x4_f32(
                /*neg_a=*/false, a, /*neg_b=*/false, b,
                /*c_mod=*/(short)0, acc[t], /*reuse_a=*/false, /*reuse_b=*/false);
        }
    }

    const int mbase = blockIdx.x * 128 + waveid * 16 + ((lane >> 4) << 3);
#pragma unroll
    for (int t = 0; t < 8; ++t) {
#pragma unroll
        for (int r = 0; r < 8; ++r)
            O[(size_t)(mbase + r) * C_DIM + t * 16 + nlane] = acc[t][r];
    }
}

// ---------------- h_new init: bias + self-loop term ----------------
__global__ void k_init_hnew(const float* __restrict__ XW, const float* __restrict__ dinv,
                            const float* __restrict__ bias, float* __restrict__ Hn, int N) {
    int idx = blockIdx.x * blockDim.x + threadIdx.x;
    if (idx < N * C_DIM) {
        int i = idx >> 7, c = idx & 127;
        float di = dinv[i];
        Hn[idx] = bias[c] + XW[idx] * di * di;   // self-loop norm = dinv[i]^2
    }
}

// ---------------- edge scatter: one wave per edge, float4 per lane ----------------
__global__ void k_scatter(const int* __restrict__ src, const int* __restrict__ dst,
                          const float* __restrict__ dinv, const float* __restrict__ XW,
                          float* __restrict__ Hn, int E) {
    int tid  = blockIdx.x * blockDim.x + threadIdx.x;
    int e    = tid >> 5;
    int lane = tid & 31;
    if (e >= E) return;
    int s = src[e], d = dst[e];
    float nrm = dinv[s] * dinv[d];
    float4 v = ((const float4*)(XW + (size_t)s * C_DIM))[lane];  // 32 lanes x 4 = 128
    float* out = Hn + (size_t)d * C_DIM + lane * 4;
    atomicAdd(out + 0, v.x * nrm);
    atomicAdd(out + 1, v.y * nrm);
    atomicAdd(out + 2, v.z * nrm);
    atomicAdd(out + 3, v.w * nrm);
}

// ---------------- traces ----------------
__global__ void k_trace_np(const float* __restrict__ H, float* __restrict__ npt, int step) {
    __shared__ float red[128];
    int k = threadIdx.x;
    red[k] = H[(size_t)k * C_DIM + k];
    __syncthreads();
    for (int s = 64; s > 0; s >>= 1) { if (k < s) red[k] += red[k + s]; __syncthreads(); }
    if (k == 0) npt[step] = red[0];
}

__global__ void k_trace_p(const float* __restrict__ H, float* __restrict__ pt, int step) {
    __shared__ float red[128];
    int b = blockIdx.x, k = threadIdx.x;
    red[k] = H[((size_t)b * 128 + k) * C_DIM + k];
    __syncthreads();
    for (int s = 64; s > 0; s >>= 1) { if (k < s) red[k] += red[k + s]; __syncthreads(); }
    if (k == 0) pt[b * WALKN + step] = red[0];
}

// ---------------- standardize + MLP head (single block, thread = batch row) ----------------
__global__ __launch_bounds__(256) void k_final(const float* __restrict__ pt,   // [B][WALK]
                                               const float* __restrict__ npt,  // [WALK]
                                               const float* __restrict__ y,    // [B]
                                               const float* __restrict__ W1,   // [WALK][15]
                                               const float* __restrict__ b1,   // [15]
                                               const float* __restrict__ W2,   // [15]
                                               const float* __restrict__ b2,   // [1]
                                               float* __restrict__ out) {      // [B]
    int b = threadIdx.x;
    float v[WALKN];
    float sgn = (y[b] - 0.5f) * 2.0f;
#pragma unroll
    for (int w = 0; w < WALKN; ++w) v[w] = (pt[b * WALKN + w] - npt[w]) * sgn;

    __shared__ float s1[256], s2[256];
    for (int w = 0; w < WALKN; ++w) {
        s1[b] = v[w];
        s2[b] = v[w] * v[w];
        __syncthreads();
        for (int s = 128; s > 0; s >>= 1) {
            if (b < s) { s1[b] += s1[b + s]; s2[b] += s2[b + s]; }
            __syncthreads();
        }
        float mean = s1[0] * (1.0f / 256.0f);
        float var  = (s2[0] - 256.0f * mean * mean) * (1.0f / 255.0f);  // ddof=1
        __syncthreads();
        v[w] = (v[w] - mean) / sqrtf(var);
    }

    float h[15];
#pragma unroll
    for (int j = 0; j < 15; ++j) {
        float acc = b1[j];
#pragma unroll
        for (int w = 0; w < WALKN; ++w) acc += v[w] * W1[w * 15 + j];
        h[j] = fmaxf(acc, 0.0f);
    }
    float o = b2[0];
#pragma unroll
    for (int j = 0; j < 15; ++j) o += h[j] * W2[j];
    out[b] = 1.0f / (1.0f + __expf(-o));
}

// ---------------- host orchestration ----------------
extern "C" void kernel_launch(void* const* d_in, const int* in_sizes, int n_in,
                              void* d_out, int out_size, void* d_ws, size_t ws_size,
                              hipStream_t stream) {
    const float* x_p  = (const float*)d_in[0];
    const float* x_np = (const float*)d_in[1];
    const float* y    = (const float*)d_in[2];
    const int*   ei_p = (const int*)d_in[3];
    const int*   ei_n = (const int*)d_in[4];
    const float* Wg   = (const float*)d_in[5];
    const float* bg   = (const float*)d_in[6];
    const float* W1   = (const float*)d_in[7];
    const float* b1   = (const float*)d_in[8];
    const float* W2   = (const float*)d_in[9];
    const float* b2   = (const float*)d_in[10];

    const int* src_p = ei_p;           const int* dst_p = ei_p + EP_N;
    const int* src_n = ei_n;           const int* dst_n = ei_n + EN_N;

    // carve workspace
    char* ws = (char*)d_ws;
    size_t off = 0;
    auto alloc = [&](size_t bytes) -> void* {
        void* p = ws + off;
        off = (off + bytes + 255) & ~(size_t)255;
        return p;
    };
    float* XW_p  = (float*)alloc((size_t)NP_N * C_DIM * sizeof(float));
    float* hA_p  = (float*)alloc((size_t)NP_N * C_DIM * sizeof(float));
    float* hB_p  = (float*)alloc((size_t)NP_N * C_DIM * sizeof(float));
    float* XW_n  = (float*)alloc((size_t)NN_N * C_DIM * sizeof(float));
    float* hA_n  = (float*)alloc((size_t)NN_N * C_DIM * sizeof(float));
    float* hB_n  = (float*)alloc((size_t)NN_N * C_DIM * sizeof(float));
    float* dinvp = (float*)alloc((size_t)NP_N * sizeof(float));
    float* dinvn = (float*)alloc((size_t)NN_N * sizeof(float));
    float* p_tr  = (float*)alloc((size_t)B_SZ * WALKN * sizeof(float));
    float* np_tr = (float*)alloc((size_t)WALKN * sizeof(float));

    // --- gcn_norm: deg -> dinv (deg starts at 1 for self loop) ---
    k_fill1<<<(NP_N + 255) / 256, 256, 0, stream>>>(dinvp, NP_N);
    k_fill1<<<1, 128, 0, stream>>>(dinvn, NN_N);
    k_degcount<<<(EP_N + 255) / 256, 256, 0, stream>>>(dst_p, dinvp, EP_N);
    k_degcount<<<(EN_N + 255) / 256, 256, 0, stream>>>(dst_n, dinvn, EN_N);
    k_rsqrt_inplace<<<(NP_N + 255) / 256, 256, 0, stream>>>(dinvp, NP_N);
    k_rsqrt_inplace<<<1, 128, 0, stream>>>(dinvn, NN_N);

    // --- walk loop ---
    const float* hin_p = x_p;   float* cur_p = hA_p; float* nxt_p = hB_p;
    const float* hin_n = x_np;  float* cur_n = hA_n; float* nxt_n = hB_n;
    for (int step = 0; step < WALKN; ++step) {
        // big (perturbed) graph
        k_gemm_wmma<<<NP_N / 128, 256, 0, stream>>>(hin_p, Wg, XW_p);
        k_init_hnew<<<(NP_N * C_DIM + 255) / 256, 256, 0, stream>>>(XW_p, dinvp, bg, cur_p, NP_N);
        k_scatter<<<(EP_N * 32) / 256, 256, 0, stream>>>(src_p, dst_p, dinvp, XW_p, cur_p, EP_N);
        k_trace_p<<<B_SZ, 128, 0, stream>>>(cur_p, p_tr, step);

        // small (non-perturbed) graph
        k_gemm_wmma<<<NN_N / 128, 256, 0, stream>>>(hin_n, Wg, XW_n);
        k_init_hnew<<<(NN_N * C_DIM + 255) / 256, 256, 0, stream>>>(XW_n, dinvn, bg, cur_n, NN_N);
        k_scatter<<<(EN_N * 32) / 256, 256, 0, stream>>>(src_n, dst_n, dinvn, XW_n, cur_n, EN_N);
        k_trace_np<<<1, 128, 0, stream>>>(cur_n, np_tr, step);

        // ping-pong
        hin_p = cur_p; { float* t = cur_p; cur_p = nxt_p; nxt_p = t; }
        hin_n = cur_n; { float* t = cur_n; cur_n = nxt_n; nxt_n = t; }
    }

    // --- head ---
    k_final<<<1, 256, 0, stream>>>(p_tr, np_tr, y, W1, b1, W2, b2, (float*)d_out);
}